// Dereverb_T60_decoder_39805756899487
// MI455X (gfx1250) — compile-verified
//
#include <hip/hip_runtime.h>

typedef __attribute__((ext_vector_type(2))) float v2f;
typedef __attribute__((ext_vector_type(8))) float v8f;

#define CIN   512
#define LIN   4000
#define KSZ   16
#define LOUT  31992           // (LIN-1)*8 - 2*8 + 16
#define UPT   15              // u-positions produced per 16-row Z tile
#define NTILE 267             // ceil(3999 / 15)

// One wave computes a 16(time) x 16(tap) tile of Z = X^T * W via
// V_WMMA_F32_16X16X4_F32, then folds Z[u+1,r] + Z[u,8+r] into the
// conv_transpose1d output through LDS.
__global__ __launch_bounds__(32)
void dereverb_t60_wmma_kernel(const float* __restrict__ X,     // (16, 512, 4000)
                              const float* __restrict__ t60,   // (8,)
                              const float* __restrict__ Wbank, // (41, 512, 1, 16)
                              float* __restrict__ out)         // (16, 1, 31992)
{
    const int tile = blockIdx.x;          // 0..266
    const int b    = blockIdx.y;          // 0..15
    const int lane = threadIdx.x;         // 0..31
    const int t0   = tile * UPT;

    __shared__ float zs[16 * 17];

    // Per-sample kernel gather: idx = round(t60[b%8]*100) - 10 (round-half-even).
    const int kidx = (int)__builtin_rintf(t60[b & 7] * 100.0f) - 10;
    const float* Wb = Wbank + (size_t)kidx * (CIN * KSZ);

    const int half = lane >> 4;           // 0: K-lanes {0,1}, 1: K-lanes {2,3}
    const int l15  = lane & 15;

    // Clamp time column; clamped rows land in Z rows that are never consumed.
    int tcol = t0 + l15;
    if (tcol > LIN - 1) tcol = LIN - 1;

    // A: X[cin, t] at b*CIN*LIN + cin*LIN + t ; lane reads cins {2h, 2h+1}
    const float* pA = X + (size_t)b * (CIN * LIN) + (size_t)(2 * half) * LIN + tcol;
    // B: W[cin, k] at cin*KSZ + k           ; lane reads cins {2h, 2h+1}, k = l15
    const float* pB = Wb + (2 * half) * KSZ + l15;

    v8f acc = {};
    #pragma unroll 4
    for (int kk = 0; kk < CIN / 4; ++kk) {
        v2f a, bb;
        a.x  = pA[0];          // A[m=l15, K=2h]
        a.y  = pA[LIN];        // A[m=l15, K=2h+1]
        bb.x = pB[0];          // B[K=2h,   n=l15]
        bb.y = pB[KSZ];        // B[K=2h+1, n=l15]
        acc = __builtin_amdgcn_wmma_f32_16x16x4_f32(
            /*neg_a=*/false, a, /*neg_b=*/false, bb,
            /*c_mod=*/(short)0, acc, /*reuse_a=*/false, /*reuse_b=*/false);
        pA += 4 * LIN;         // next 4 input channels
        pB += 4 * KSZ;
    }

    // Spill Z tile to LDS. Lane holds Z[v + 8*half][l15] in acc[v].
    #pragma unroll
    for (int v = 0; v < 8; ++v)
        zs[(v + 8 * half) * 17 + l15] = acc[v];
    __syncthreads();

    // y[b, 8u+r] = Z[u+1, r] + Z[u, 8+r],  u = t0 .. t0+14 (guarded at 3998)
    float* outb = out + (size_t)b * LOUT;
    #pragma unroll
    for (int i = lane; i < UPT * 8; i += 32) {
        const int mu = i >> 3;
        const int r  = i & 7;
        const int u  = t0 + mu;
        if (u < LIN - 1)
            outb[u * 8 + r] = zs[(mu + 1) * 17 + r] + zs[mu * 17 + 8 + r];
    }
}

extern "C" void kernel_launch(void* const* d_in, const int* in_sizes, int n_in,
                              void* d_out, int out_size, void* d_ws, size_t ws_size,
                              hipStream_t stream) {
    const float* X     = (const float*)d_in[0];   // (16, 512, 4000) f32
    const float* t60   = (const float*)d_in[1];   // (8,) f32
    const float* Wbank = (const float*)d_in[2];   // (41, 512, 1, 16) f32
    float* out         = (float*)d_out;           // (16, 1, 31992) f32
    (void)in_sizes; (void)n_in; (void)out_size; (void)d_ws; (void)ws_size;

    dim3 grid(NTILE, 16);
    dim3 block(32);
    dereverb_t60_wmma_kernel<<<grid, block, 0, stream>>>(X, t60, Wbank, out);
}